// ContinuousGaussianVFE_81346680586526
// MI455X (gfx1250) — compile-verified
//
#include <hip/hip_runtime.h>
#include <hip/hip_bf16.h>
#include <stdint.h>

// ---------------------------------------------------------------------------
// ContinuousGaussianVFE for MI455X (gfx1250).
// Memory-bound pipeline; unique() realized as a 4MB L2-resident bitmap over
// the 2^25 code space + popcount prefix scan (gives sorted-unique ranks).
// Feature accumulation double-buffers 16KB tiles in LDS via gfx1250
// async-to-LDS ops with partial ASYNCcnt waits (DMA overlapped with scatter).
// ---------------------------------------------------------------------------

#define AS1 __attribute__((address_space(1)))
#define AS3 __attribute__((address_space(3)))

typedef int v4i __attribute__((vector_size(16)));

namespace {
constexpr int   kN         = 1000000;
constexpr int   SCALE_XYZ  = 512 * 512 * 32;   // 8388608
constexpr int   SCALE_YZ   = 512 * 32;         // 16384
constexpr int   SCALE_Z    = 32;
constexpr int   kCodeSpace = 4 * SCALE_XYZ;    // 2^25
constexpr int   kNW        = kCodeSpace / 32;  // 2^20 bitmap words
constexpr int   kTiles     = (kN + 63) / 64;   // 15625 feature tiles of 64 pts
constexpr int   kTPB       = 4;                // tiles pipelined per block
}

// ---- gfx1250 async-to-LDS path (builtin confirmed on this toolchain) -------
#if defined(__has_builtin) && __has_builtin(__builtin_amdgcn_global_load_async_to_lds_b128)
#define ASYNC_LD16(gsrc, ldst)                                                   \
  __builtin_amdgcn_global_load_async_to_lds_b128((AS1 v4i*)(gsrc),               \
                                                 (AS3 v4i*)(ldst), 0, 0)
#else
#define ASYNC_LD16(gsrc, ldst)                                                   \
  do {                                                                           \
    unsigned _lofs = (unsigned)(__SIZE_TYPE__)(AS3 void*)(ldst);                 \
    const void* _g = (const void*)(gsrc);                                        \
    asm volatile("global_load_async_to_lds_b128 %0, %1, off"                     \
                 :: "v"(_lofs), "v"(_g) : "memory");                             \
  } while (0)
#endif

#if defined(__has_builtin) && __has_builtin(__builtin_amdgcn_s_wait_asynccnt)
#define WAIT_ASYNC(n) __builtin_amdgcn_s_wait_asynccnt(n)
#else
#define WAIT_ASYNC(n) asm volatile("s_wait_asynccnt %0" :: "i"(n) : "memory")
#endif

// ---------------------------------------------------------------------------
// K0: zero outputs + workspace accumulators.
__global__ void k_init(float* __restrict__ out, int* __restrict__ counts,
                       int* __restrict__ refIdx, unsigned* __restrict__ bitmap) {
  const long long stride = (long long)gridDim.x * blockDim.x;
  const long long t0 = (long long)blockIdx.x * blockDim.x + threadIdx.x;
  const long long totOut = 94LL * kN;
  for (long long i = t0; i < totOut; i += stride) out[i] = 0.0f;
  for (long long i = t0; i < kN; i += stride) { counts[i] = 0; refIdx[i] = 0x7fffffff; }
  for (long long i = t0; i < kNW; i += stride) bitmap[i] = 0u;
}

// K1: voxel code per point + mark presence bitmap.
__global__ void k_mark(const float* __restrict__ mu, const int* __restrict__ batch,
                       int* __restrict__ codes, unsigned* __restrict__ bitmap) {
  int i = blockIdx.x * blockDim.x + threadIdx.x;
  if (i >= kN) return;
  float x = mu[i * 3 + 0], y = mu[i * 3 + 1], z = mu[i * 3 + 2];
  int ix = (int)floorf((x - (-51.2f)) / 0.2f);
  int iy = (int)floorf((y - (-51.2f)) / 0.2f);
  int iz = (int)floorf((z - (-3.2f)) / 0.2f);
  int code = batch[i] * SCALE_XYZ + iz * SCALE_YZ + iy * SCALE_Z + ix;
  code = min(max(code, 0), kCodeSpace - 1);
  codes[i] = code;
  atomicOr(&bitmap[code >> 5], 1u << (code & 31));
}

// K2: per-word popcount exclusive scan (block-local) + block sums.
__global__ void k_scan_words(const unsigned* __restrict__ bitmap,
                             int* __restrict__ wordPrefix, int* __restrict__ blockSums) {
  __shared__ int sh[256];
  const int t = threadIdx.x;
  const int base = blockIdx.x * 1024 + t * 4;
  uint4 wv = ((const uint4*)bitmap)[base >> 2];
  int p0 = __popc(wv.x), p1 = __popc(wv.y), p2 = __popc(wv.z), p3 = __popc(wv.w);
  int own = p0 + p1 + p2 + p3;
  sh[t] = own;
  __syncthreads();
  for (int off = 1; off < 256; off <<= 1) {
    int v = sh[t];
    if (t >= off) v += sh[t - off];
    __syncthreads();
    sh[t] = v;
    __syncthreads();
  }
  int excl = sh[t] - own;
  wordPrefix[base + 0] = excl;
  wordPrefix[base + 1] = excl + p0;
  wordPrefix[base + 2] = excl + p0 + p1;
  wordPrefix[base + 3] = excl + p0 + p1 + p2;
  if (t == 255) blockSums[blockIdx.x] = sh[255];
}

// K3: scan 1024 block sums (single block), emit total unique count M.
__global__ void k_scan_blocks(const int* __restrict__ blockSums,
                              int* __restrict__ blockOffs, int* __restrict__ Mptr) {
  __shared__ int sh[1024];
  const int t = threadIdx.x;
  int own = blockSums[t];
  sh[t] = own;
  __syncthreads();
  for (int off = 1; off < 1024; off <<= 1) {
    int v = sh[t];
    if (t >= off) v += sh[t - off];
    __syncthreads();
    sh[t] = v;
    __syncthreads();
  }
  blockOffs[t] = sh[t] - own;
  if (t == 1023) *Mptr = sh[1023];
}

// K4: globalize the word prefix.
__global__ void k_apply(int* __restrict__ wordPrefix, const int* __restrict__ blockOffs) {
  int w = blockIdx.x * blockDim.x + threadIdx.x;
  if (w < kNW) wordPrefix[w] += blockOffs[w >> 10];
}

// K5: inv[i] = sorted-unique rank; accumulate counts/refMin/mu/scale/semantic.
__global__ void k_point(const float* __restrict__ mu, const float* __restrict__ scale,
                        const float* __restrict__ sem, const int* __restrict__ codes,
                        const unsigned* __restrict__ bitmap, const int* __restrict__ wordPrefix,
                        int* __restrict__ inv, int* __restrict__ counts, int* __restrict__ refIdx,
                        float* __restrict__ muOut, float* __restrict__ scOut,
                        float* __restrict__ semOut) {
  int i = blockIdx.x * blockDim.x + threadIdx.x;
  if (i >= kN) return;
  int code = codes[i];
  int w = code >> 5, b = code & 31;
  int seg = wordPrefix[w] + __popc(bitmap[w] & ((1u << b) - 1u));
  inv[i] = seg;
  atomicAdd(&counts[seg], 1);
  atomicMin(&refIdx[seg], i);
#pragma unroll
  for (int j = 0; j < 3; ++j) {
    atomicAdd(&muOut[seg * 3 + j], mu[i * 3 + j]);
    atomicAdd(&scOut[seg * 3 + j], scale[i * 3 + j]);
  }
#pragma unroll
  for (int c = 0; c < 16; ++c)
    atomicAdd(&semOut[seg * 16 + c], sem[i * 16 + c]);
}

// K6: antipodal quaternion accumulation against each segment's first point.
__global__ void k_quat(const float* __restrict__ rot, const int* __restrict__ inv,
                       const int* __restrict__ refIdx, float* __restrict__ rotOut) {
  int i = blockIdx.x * blockDim.x + threadIdx.x;
  if (i >= kN) return;
  int seg = inv[i];
  int r = refIdx[seg];
  float4 q = ((const float4*)rot)[i];
  float4 f = ((const float4*)rot)[r];
  float dot = q.x * f.x + q.y * f.y + q.z * f.z + q.w * f.w;
  float t = dot + 1e-8f;
  float s = (t > 0.0f) ? 1.0f : ((t < 0.0f) ? -1.0f : 0.0f);
  atomicAdd(&rotOut[seg * 4 + 0], q.x * s);
  atomicAdd(&rotOut[seg * 4 + 1], q.y * s);
  atomicAdd(&rotOut[seg * 4 + 2], q.z * s);
  atomicAdd(&rotOut[seg * 4 + 3], q.w * s);
}

// K7: feature scatter with double-buffered async-to-LDS pipeline.
// Each wave issues 4 async loads per 16KB tile; while tile t is scattered,
// tile t+1's DMA is in flight (s_wait_asynccnt 4 retires only tile t's loads).
__global__ void k_feat(const float* __restrict__ features, const int* __restrict__ inv,
                       float* __restrict__ featOut) {
  __shared__ float tile[2][64 * 64];   // 2 x 16 KB double buffer
  const int t = threadIdx.x;
  const int t0 = blockIdx.x * kTPB;

  {  // prologue: issue DMA for first tile (every block owns >= 1 valid tile)
    const float* src = features + (size_t)t0 * 64 * 64;
#pragma unroll
    for (int k = 0; k < 4; ++k) {
      int f = (k * 256 + t) * 4;       // 16B chunk per lane
      ASYNC_LD16(src + f, &tile[0][f]);
    }
  }

  for (int ti = 0; ti < kTPB; ++ti) {
    const int tileIdx = t0 + ti;
    if (tileIdx >= kTiles) break;
    const int buf = ti & 1;
    const bool hasNext = (ti + 1 < kTPB) && (tileIdx + 1 < kTiles);
    if (hasNext) {
      const float* nsrc = features + (size_t)(tileIdx + 1) * 64 * 64;
#pragma unroll
      for (int k = 0; k < 4; ++k) {
        int f = (k * 256 + t) * 4;
        ASYNC_LD16(nsrc + f, &tile[buf ^ 1][f]);
      }
      if (tileIdx + 2 < kTiles)        // prime tile t+2 in L2 while DMAs run
        __builtin_prefetch(nsrc + 64 * 64 + t * 16, 0, 1);
      WAIT_ASYNC(4);                   // tile t done; tile t+1 stays in flight
    } else {
      WAIT_ASYNC(0);
    }
    __syncthreads();                   // tile t visible to all waves

    const int p = t >> 2;              // point 0..63 within tile
    const int strip = (t & 3) * 16;    // channel strip
    const int seg = inv[tileIdx * 64 + p];
    float* dst = featOut + (size_t)seg * 64 + strip;
#pragma unroll
    for (int c = 0; c < 16; ++c)
      atomicAdd(&dst[c], tile[buf][p * 64 + strip + c]);
    __syncthreads();                   // buffer reusable for tile t+2
  }
}

// K8: per-word finalize: divide sums by count, normalize quat, decode coords.
__global__ void k_fin_seg(const unsigned* __restrict__ bitmap, const int* __restrict__ wordPrefix,
                          const int* __restrict__ counts, float* __restrict__ muOut,
                          float* __restrict__ scOut, float* __restrict__ rotOut,
                          float* __restrict__ featOut, float* __restrict__ semOut,
                          int* __restrict__ vcOut) {
  int w = blockIdx.x * blockDim.x + threadIdx.x;
  if (w >= kNW) return;
  unsigned bits = bitmap[w];
  if (!bits) return;
  int seg = wordPrefix[w];
  unsigned m = bits;
  while (m) {
    int bit = __builtin_ctz(m);
    m &= m - 1u;
    int code = (w << 5) | bit;
    int vb = code / SCALE_XYZ; int rem = code % SCALE_XYZ;
    int vz = rem / SCALE_YZ;   rem %= SCALE_YZ;
    int vy = rem / SCALE_Z;    int vx = rem % SCALE_Z;
    vcOut[seg * 4 + 0] = vb; vcOut[seg * 4 + 1] = vz;
    vcOut[seg * 4 + 2] = vy; vcOut[seg * 4 + 3] = vx;
    float d = fmaxf((float)counts[seg], 1.0f);
    float id = 1.0f / d;
#pragma unroll
    for (int j = 0; j < 3; ++j) { muOut[seg * 3 + j] *= id; scOut[seg * 3 + j] *= id; }
#pragma unroll
    for (int c = 0; c < 16; ++c) semOut[seg * 16 + c] *= id;
    for (int c = 0; c < 64; ++c) featOut[seg * 64 + c] *= id;
    float q0 = rotOut[seg * 4 + 0] * id, q1 = rotOut[seg * 4 + 1] * id;
    float q2 = rotOut[seg * 4 + 2] * id, q3 = rotOut[seg * 4 + 3] * id;
    float n = fmaxf(sqrtf(q0 * q0 + q1 * q1 + q2 * q2 + q3 * q3), 1e-12f);
    rotOut[seg * 4 + 0] = q0 / n; rotOut[seg * 4 + 1] = q1 / n;
    rotOut[seg * 4 + 2] = q2 / n; rotOut[seg * 4 + 3] = q3 / n;
    ++seg;
  }
}

// K9: padded segments (unq == -1): coords decode to (-1, 2047, 511, 31).
__global__ void k_fin_pad(const int* __restrict__ Mptr, int* __restrict__ vcOut) {
  const int M = *Mptr;
  const int stride = gridDim.x * blockDim.x;
  for (int s = blockIdx.x * blockDim.x + threadIdx.x; s < kN; s += stride) {
    if (s >= M) {
      vcOut[s * 4 + 0] = -1;  vcOut[s * 4 + 1] = 2047;
      vcOut[s * 4 + 2] = 511; vcOut[s * 4 + 3] = 31;
    }
  }
}

extern "C" void kernel_launch(void* const* d_in, const int* in_sizes, int n_in,
                              void* d_out, int out_size, void* d_ws, size_t ws_size,
                              hipStream_t stream) {
  (void)in_sizes; (void)n_in; (void)out_size; (void)ws_size;
  const float* mu    = (const float*)d_in[0];
  const float* scale = (const float*)d_in[1];
  const float* rot   = (const float*)d_in[2];
  const float* feat  = (const float*)d_in[3];
  const float* sem   = (const float*)d_in[4];
  const int*   batch = (const int*)d_in[5];

  float* out     = (float*)d_out;
  float* muOut   = out;
  float* scOut   = out + 3LL * kN;
  float* rotOut  = out + 6LL * kN;
  float* featOut = out + 10LL * kN;
  float* semOut  = out + 74LL * kN;
  int*   vcOut   = (int*)(out + 90LL * kN);

  int* ws = (int*)d_ws;
  int*      codes      = ws;              ws += kN;
  int*      inv        = ws;              ws += kN;
  int*      counts     = ws;              ws += kN;
  int*      refIdx     = ws;              ws += kN;
  unsigned* bitmap     = (unsigned*)ws;   ws += kNW;
  int*      wordPrefix = ws;              ws += kNW;
  int*      blockSums  = ws;              ws += 1024;
  int*      blockOffs  = ws;              ws += 1024;
  int*      Mptr       = ws;              ws += 1;

  const int nb = (kN + 255) / 256;
  const int featBlocks = (kTiles + kTPB - 1) / kTPB;
  k_init<<<2048, 256, 0, stream>>>(out, counts, refIdx, bitmap);
  k_mark<<<nb, 256, 0, stream>>>(mu, batch, codes, bitmap);
  k_scan_words<<<1024, 256, 0, stream>>>(bitmap, wordPrefix, blockSums);
  k_scan_blocks<<<1, 1024, 0, stream>>>(blockSums, blockOffs, Mptr);
  k_apply<<<kNW / 256, 256, 0, stream>>>(wordPrefix, blockOffs);
  k_point<<<nb, 256, 0, stream>>>(mu, scale, sem, codes, bitmap, wordPrefix,
                                  inv, counts, refIdx, muOut, scOut, semOut);
  k_quat<<<nb, 256, 0, stream>>>(rot, inv, refIdx, rotOut);
  k_feat<<<featBlocks, 256, 0, stream>>>(feat, inv, featOut);
  k_fin_seg<<<kNW / 256, 256, 0, stream>>>(bitmap, wordPrefix, counts, muOut, scOut,
                                           rotOut, featOut, semOut, vcOut);
  k_fin_pad<<<2048, 256, 0, stream>>>(Mptr, vcOut);
}